// SparseBAGE_36601711296804
// MI455X (gfx1250) — compile-verified
//
#include <hip/hip_runtime.h>
#include <hip/hip_bf16.h>

// GCN: 3x (GEMM -> graph aggregate) with BN+ReLU between layers.
// N=50000 nodes, E=600000 edges, DIN=H=128, DOUT=40.

typedef __attribute__((ext_vector_type(2))) float v2f;
typedef __attribute__((ext_vector_type(8))) float v8f;

#define BN_EPS 1e-5f

static inline int cdiv(long long a, long long b) { return (int)((a + b - 1) / b); }

// ---------------- utility kernels ----------------

__global__ void fill_f32(float* __restrict__ p, float v, int n) {
  int i = blockIdx.x * blockDim.x + threadIdx.x;
  if (i < n) p[i] = v;
}

__global__ void deg_scatter(const int* __restrict__ col, float* __restrict__ deg, int E) {
  int e = blockIdx.x * blockDim.x + threadIdx.x;
  if (e < E) atomicAdd(&deg[col[e]], 1.0f);
}

__global__ void deg_to_dinv(float* __restrict__ d, int n) {
  int i = blockIdx.x * blockDim.x + threadIdx.x;
  if (i < n) d[i] = rsqrtf(d[i]);   // deg >= 1 always (self-loops)
}

// ---------------- WMMA GEMM: Y[nrows x NT*16] = X[nrows x 128] * W[128 x NT*16] ----------------
// fp32 WMMA (V_WMMA_F32_16X16X4_F32). One wave computes a 16-row strip over all
// NT column tiles; W staged in LDS per workgroup. 256 threads = 8 waves = 128 rows/block.

template <int NT>
__global__ __launch_bounds__(256) void gemm_wmma_f32(const float* __restrict__ X,
                                                     const float* __restrict__ W,
                                                     float* __restrict__ Y, int nrows) {
  constexpr int LDW = NT * 16;
  __shared__ float sW[128 * LDW];
  const int tid = threadIdx.x;
  for (int i = tid; i < 128 * LDW; i += 256) sW[i] = W[i];
  __syncthreads();

  const int wave = tid >> 5;
  const int lane = tid & 31;
  const int m0 = (blockIdx.x * 8 + wave) * 16;
  if (m0 >= nrows) return;                    // wave-uniform: EXEC stays all-1s for WMMA

  const int lrow = lane & 15;                 // row-within-tile for A; col for B/C/D
  const int hi = lane >> 4;                   // lane half selects K pair / row half
  int arow = m0 + lrow;
  if (arow >= nrows) arow = nrows - 1;        // clamp loads; stores guarded below
  const float* __restrict__ xrow = X + (size_t)arow * 128;

  v8f acc[NT] = {};
  for (int kb = 0; kb < 32; ++kb) {           // K = 128 = 32 * 4
    const int k0 = kb * 4 + hi * 2;
    v2f a;
    a.x = xrow[k0 + 0];                       // A 16x4 f32: V0=K0/K2, V1=K1/K3
    a.y = xrow[k0 + 1];
#pragma unroll
    for (int t = 0; t < NT; ++t) {
      v2f b;
      b.x = sW[(k0 + 0) * LDW + t * 16 + lrow];  // B 4x16 f32: row striped across lanes
      b.y = sW[(k0 + 1) * LDW + t * 16 + lrow];
      acc[t] = __builtin_amdgcn_wmma_f32_16x16x4_f32(
          /*neg_a=*/false, a, /*neg_b=*/false, b,
          /*c_mod=*/(short)0, acc[t], /*reuse_a=*/false, /*reuse_b=*/false);
    }
  }
  // C/D layout: VGPR r -> (M = r + 8*hi, N = lrow)
#pragma unroll
  for (int t = 0; t < NT; ++t) {
#pragma unroll
    for (int r = 0; r < 8; ++r) {
      int m = m0 + r + hi * 8;
      if (m < nrows) Y[(size_t)m * LDW + t * 16 + lrow] = acc[t][r];
    }
  }
}

// ---------------- graph aggregation ----------------
// agg[i,:] = dinv[i]^2 * hlin[i,:] + bias  (self-loop + bias)
__global__ void agg_init(const float* __restrict__ hlin, const float* __restrict__ dinv,
                         const float* __restrict__ bias, float* __restrict__ agg,
                         int n, int ho, int nb) {
  long long i = (long long)blockIdx.x * blockDim.x + threadIdx.x;
  if (i >= (long long)n * ho) return;
  int node = (int)(i / ho);
  int f = (int)(i % ho);
  float d = dinv[node];
  float bb = (f < nb) ? bias[f] : 0.0f;
  agg[i] = d * d * hlin[i] + bb;
}

// agg[col[e],:] += dinv[row]*dinv[col] * hlin[row[e],:]  — 1 float4 chunk per thread,
// `chunks` consecutive threads share one edge (coalesced source-row read).
__global__ void agg_scatter(const int* __restrict__ row, const int* __restrict__ col,
                            const float* __restrict__ dinv, const float* __restrict__ hlin,
                            float* __restrict__ agg, int E, int chunks, int ld) {
  long long t = (long long)blockIdx.x * blockDim.x + threadIdx.x;
  if (t >= (long long)E * chunks) return;
  int e = (int)(t / chunks);
  int c = (int)(t % chunks);
  int r = row[e], cl = col[e];
  float nrm = dinv[r] * dinv[cl];
  const float4 v = *reinterpret_cast<const float4*>(hlin + (size_t)r * ld + c * 4);
  float* dst = agg + (size_t)cl * ld + c * 4;
  atomicAdd(dst + 0, nrm * v.x);
  atomicAdd(dst + 1, nrm * v.y);
  atomicAdd(dst + 2, nrm * v.z);
  atomicAdd(dst + 3, nrm * v.w);
}

// ---------------- batch norm (feature dim = 128) ----------------
__global__ __launch_bounds__(128) void bn_stats(const float* __restrict__ h,
                                                float* __restrict__ sums, int n) {
  int f = threadIdx.x;                        // one feature column per thread
  float s = 0.0f, s2 = 0.0f;
  for (int r = blockIdx.x; r < n; r += gridDim.x) {
    float v = h[(size_t)r * 128 + f];
    s += v;
    s2 += v * v;
  }
  atomicAdd(&sums[f], s);
  atomicAdd(&sums[128 + f], s2);
}

__global__ void bn_apply_relu(const float* __restrict__ agg, const float* __restrict__ sums,
                              const float* __restrict__ g, const float* __restrict__ be,
                              float* __restrict__ out, int n, float invN) {
  long long i = (long long)blockIdx.x * blockDim.x + threadIdx.x;
  if (i >= (long long)n * 128) return;
  int f = (int)(i & 127);
  float mean = sums[f] * invN;
  float var = sums[128 + f] * invN - mean * mean;  // biased var (matches reference)
  float v = (agg[i] - mean) * rsqrtf(var + BN_EPS) * g[f] + be[f];
  out[i] = v > 0.0f ? v : 0.0f;
}

// ---------------- layer-3 helpers ----------------
__global__ void pad_w3(const float* __restrict__ W3, float* __restrict__ Wp) {
  int i = blockIdx.x * blockDim.x + threadIdx.x;   // 128*48
  if (i >= 128 * 48) return;
  int k = i / 48, nn = i % 48;
  Wp[i] = (nn < 40) ? W3[k * 40 + nn] : 0.0f;
}

__global__ void copy_out(const float* __restrict__ agg, float* __restrict__ out, int n) {
  long long i = (long long)blockIdx.x * blockDim.x + threadIdx.x;  // n*40
  if (i >= (long long)n * 40) return;
  int node = (int)(i / 40), f = (int)(i % 40);
  float v = agg[(size_t)node * 48 + f];
  out[i] = v;                                      // tuple (out, out): both halves
  out[(long long)n * 40 + i] = v;
}

// ---------------- driver ----------------

extern "C" void kernel_launch(void* const* d_in, const int* in_sizes, int n_in,
                              void* d_out, int out_size, void* d_ws, size_t ws_size,
                              hipStream_t stream) {
  const float* x  = (const float*)d_in[0];
  const int*   ei = (const int*)d_in[1];
  const float* W1 = (const float*)d_in[2];
  const float* b1 = (const float*)d_in[3];
  const float* W2 = (const float*)d_in[4];
  const float* b2 = (const float*)d_in[5];
  const float* W3 = (const float*)d_in[6];
  const float* b3 = (const float*)d_in[7];
  const float* g1 = (const float*)d_in[8];
  const float* be1 = (const float*)d_in[9];
  const float* g2 = (const float*)d_in[10];
  const float* be2 = (const float*)d_in[11];

  const int N = in_sizes[0] / 128;
  const int E = in_sizes[1] / 2;
  const int* erow = ei;
  const int* ecol = ei + E;
  const float invN = 1.0f / (float)N;
  const size_t NH = (size_t)N * 128;

  // workspace carve (256B aligned slices)
  char* ws = (char*)d_ws;
  size_t off = 0;
  auto carve = [&](size_t bytes) {
    float* p = (float*)(ws + off);
    off += (bytes + 255) & ~(size_t)255;
    return p;
  };
  float* dinv = carve((size_t)N * 4);
  float* bufA = carve(NH * 4);      // GEMM output (h_lin)
  float* bufB = carve(NH * 4);      // aggregated
  float* bufC = carve(NH * 4);      // post BN+ReLU
  float* sums = carve(256 * 4);     // [sum(128) | sumsq(128)]
  float* Wp   = carve(128 * 48 * 4);

  const int T = 256;
  const int gemmBlocks = cdiv(N, 128);

  // ---- degree / norm ----
  fill_f32<<<cdiv(N, T), T, 0, stream>>>(dinv, 1.0f, N);           // self-loop
  deg_scatter<<<cdiv(E, T), T, 0, stream>>>(ecol, dinv, E);
  deg_to_dinv<<<cdiv(N, T), T, 0, stream>>>(dinv, N);

  // ---- layer 1 ----
  gemm_wmma_f32<8><<<gemmBlocks, T, 0, stream>>>(x, W1, bufA, N);
  agg_init<<<cdiv(NH, T), T, 0, stream>>>(bufA, dinv, b1, bufB, N, 128, 128);
  agg_scatter<<<cdiv((long long)E * 32, T), T, 0, stream>>>(erow, ecol, dinv, bufA, bufB, E, 32, 128);
  fill_f32<<<1, 256, 0, stream>>>(sums, 0.0f, 256);
  bn_stats<<<512, 128, 0, stream>>>(bufB, sums, N);
  bn_apply_relu<<<cdiv(NH, T), T, 0, stream>>>(bufB, sums, g1, be1, bufC, N, invN);

  // ---- layer 2 ----
  gemm_wmma_f32<8><<<gemmBlocks, T, 0, stream>>>(bufC, W2, bufA, N);
  agg_init<<<cdiv(NH, T), T, 0, stream>>>(bufA, dinv, b2, bufB, N, 128, 128);
  agg_scatter<<<cdiv((long long)E * 32, T), T, 0, stream>>>(erow, ecol, dinv, bufA, bufB, E, 32, 128);
  fill_f32<<<1, 256, 0, stream>>>(sums, 0.0f, 256);
  bn_stats<<<512, 128, 0, stream>>>(bufB, sums, N);
  bn_apply_relu<<<cdiv(NH, T), T, 0, stream>>>(bufB, sums, g2, be2, bufC, N, invN);

  // ---- layer 3 (pad 40 -> 48 cols so WMMA tiles + float4 scatter apply) ----
  pad_w3<<<cdiv(128 * 48, T), T, 0, stream>>>(W3, Wp);
  gemm_wmma_f32<3><<<gemmBlocks, T, 0, stream>>>(bufC, Wp, bufA, N);
  agg_init<<<cdiv((long long)N * 48, T), T, 0, stream>>>(bufA, dinv, b3, bufB, N, 48, 40);
  agg_scatter<<<cdiv((long long)E * 12, T), T, 0, stream>>>(erow, ecol, dinv, bufA, bufB, E, 12, 48);
  copy_out<<<cdiv((long long)N * 40, T), T, 0, stream>>>(bufB, (float*)d_out, N);
}